// floss_30751965839532
// MI455X (gfx1250) — compile-verified
//
#include <hip/hip_runtime.h>

#define WDIM   224
#define WW     (WDIM * WDIM)       // 50176
#define NB     256                 // batch
#define TPB    256
#define CHUNKS 7                   // chunks per sample in main pass
#define ITERS  7                   // float4 iters per thread per chunk: 7*256*4 = 7168
#define CHUNK_ELEMS (ITERS * TPB * 4)
#define NBLK2  (NB * CHUNKS)       // 1792 partial sums
#define K1_ITERS 49                // 49*256*4 = 50176 elems per sample

typedef __attribute__((ext_vector_type(2))) float v2f;
typedef __attribute__((ext_vector_type(8))) float v8f;

// ---------------------------------------------------------------------------
// Pass 1: per-sample max + argmax centroid in ONE read of target.
// Per-thread tuple (max, cnt, sum_i, sum_j), merged via LDS tree.
// Integer accumulators: sum_i <= 50176*223 ~ 11.2M < 2^32, exact.
// ---------------------------------------------------------------------------
__global__ __launch_bounds__(TPB)
void stats_kernel(const float* __restrict__ target, float* __restrict__ stats)
{
    const int b   = blockIdx.x;
    const int tid = threadIdx.x;
    const float* t = target + (size_t)b * WW;

    float    m   = -3.402823466e+38f;
    unsigned cnt = 0u, si = 0u, sj = 0u;

    for (int it = 0; it < K1_ITERS; ++it) {
        const int e = (it * TPB + tid) * 4;
        const float4 v4 = *reinterpret_cast<const float4*>(t + e);
        const int i = e / WDIM;            // row (axis 1)
        const int j = e - i * WDIM;        // col (axis 2); float4 never crosses a row
        const float vv[4] = {v4.x, v4.y, v4.z, v4.w};
#pragma unroll
        for (int k = 0; k < 4; ++k) {
            const float v = vv[k];
            if (v > m)       { m = v; cnt = 1u; si = (unsigned)i; sj = (unsigned)(j + k); }
            else if (v == m) { cnt++; si += (unsigned)i; sj += (unsigned)(j + k); }
        }
    }

    __shared__ float    sm[TPB];
    __shared__ unsigned sc[TPB], sri[TPB], srj[TPB];
    sm[tid] = m; sc[tid] = cnt; sri[tid] = si; srj[tid] = sj;
    __syncthreads();
    for (int stride = TPB / 2; stride > 0; stride >>= 1) {
        if (tid < stride) {
            const float mo = sm[tid + stride];
            if (mo > sm[tid]) {
                sm[tid] = mo; sc[tid] = sc[tid + stride];
                sri[tid] = sri[tid + stride]; srj[tid] = srj[tid + stride];
            } else if (mo == sm[tid]) {
                sc[tid]  += sc[tid + stride];
                sri[tid] += sri[tid + stride];
                srj[tid] += srj[tid + stride];
            }
        }
        __syncthreads();
    }
    if (tid == 0) {
        const float c = (float)sc[0];
        stats[2 * b + 0] = (float)sri[0] / c;   // x: mean row index
        stats[2 * b + 1] = (float)srj[0] / c;   // y: mean col index
    }
}

// ---------------------------------------------------------------------------
// Pass 2: streaming fused weight*BCE + block reduction (bandwidth-bound).
// 1792 blocks (7 chunks/sample); float4 loads; 4 trans ops per element.
// ---------------------------------------------------------------------------
__global__ __launch_bounds__(TPB)
void main_kernel(const float* __restrict__ input, const float* __restrict__ target,
                 const float* __restrict__ stats, float* __restrict__ partials)
{
    const int blk   = blockIdx.x;
    const int b     = blk / CHUNKS;
    const int chunk = blk - b * CHUNKS;
    const int tid   = threadIdx.x;
    const size_t base = (size_t)b * WW;
    const float x = stats[2 * b + 0];
    const float y = stats[2 * b + 1];

    float acc = 0.0f;
#pragma unroll
    for (int it = 0; it < ITERS; ++it) {
        const int e = chunk * CHUNK_ELEMS + (it * TPB + tid) * 4;
        const float4 p4 = *reinterpret_cast<const float4*>(input + base + e);
        const float4 t4 = *reinterpret_cast<const float4*>(target + base + e);
        const int i = e / WDIM;
        const int j = e - i * WDIM;
        const float di = (float)i - x;
        const float pp[4] = {p4.x, p4.y, p4.z, p4.w};
        const float tt[4] = {t4.x, t4.y, t4.z, t4.w};
#pragma unroll
        for (int k = 0; k < 4; ++k) {
            const float dj = (float)(j + k) - y;
            const float d  = sqrtf(__builtin_fmaf(di, di, dj * dj));  // v_sqrt_f32
            const float w  = (float)WDIM / (d + 1.0f);                // v_rcp_f32
            const float p  = pp[k], t = tt[k];
            const float lp = fmaxf(__logf(p),        -100.0f);        // v_log_f32
            const float l1 = fmaxf(__logf(1.0f - p), -100.0f);        // v_log_f32
            const float bce = -__builtin_fmaf(t, lp - l1, l1);        // t*lp + (1-t)*l1
            acc = __builtin_fmaf(w, bce, acc);
        }
    }

    __shared__ float red[TPB];
    red[tid] = acc;
    __syncthreads();
    for (int stride = TPB / 2; stride > 0; stride >>= 1) {
        if (tid < stride) red[tid] += red[tid + stride];
        __syncthreads();
    }
    if (tid == 0) partials[blk] = red[0];
}

// ---------------------------------------------------------------------------
// Pass 3: deterministic final reduce. 1792 -> 256 by strided sums, then
// wave 0 reduces 256 floats with a chain of 4x V_WMMA_F32_16X16X4_F32:
// A = ones(16x4) so D[m,n] = colsum(B)[n] + C[m,n]; all D rows identical,
// so the 32-lane sum of D's first VGPR equals 2x the grand total.
// ---------------------------------------------------------------------------
__global__ __launch_bounds__(TPB)
void finalize_kernel(const float* __restrict__ partials, float* __restrict__ out)
{
    __shared__ float s[TPB];
    const int tid = threadIdx.x;
    float a = 0.0f;
#pragma unroll
    for (int q = 0; q < CHUNKS; ++q) a += partials[tid + TPB * q];
    s[tid] = a;
    __syncthreads();

    if (tid < 32) {                       // whole wave 0 active: EXEC all ones
        const int lane = tid;
        v2f A; A[0] = 1.0f; A[1] = 1.0f;  // ones(16x4) regardless of layout
        v8f C = {};
#pragma unroll
        for (int q = 0; q < 4; ++q) {
            v2f Bm;
            Bm[0] = s[lane + 64 * q];
            Bm[1] = s[lane + 32 + 64 * q];
            C = __builtin_amdgcn_wmma_f32_16x16x4_f32(
                    false, A, false, Bm, (short)0, C, false, false);
        }
        float v = C[0];                   // row M=0 / M=8 across the wave
#pragma unroll
        for (int off = 16; off > 0; off >>= 1) v += __shfl_xor(v, off, 32);
        if (lane == 0)
            out[0] = v * 0.5f * (1.0f / ((float)NB * (float)WW));
    }
}

extern "C" void kernel_launch(void* const* d_in, const int* in_sizes, int n_in,
                              void* d_out, int out_size, void* d_ws, size_t ws_size,
                              hipStream_t stream)
{
    const float* input  = (const float*)d_in[0];
    const float* target = (const float*)d_in[1];
    float* stats    = (float*)d_ws;          // 512 floats: (x,y) per sample
    float* partials = stats + 2 * NB;        // 1792 floats: per-block partial sums
    float* outp     = (float*)d_out;

    stats_kernel   <<<NB,    TPB, 0, stream>>>(target, stats);
    main_kernel    <<<NBLK2, TPB, 0, stream>>>(input, target, stats, partials);
    finalize_kernel<<<1,     TPB, 0, stream>>>(partials, outp);
}